// AttWModel_206158430484
// MI455X (gfx1250) — compile-verified
//
#include <hip/hip_runtime.h>
#include <math.h>

#define DIM   128
#define TL    64
#define PADW  (DIM + 4)              // 132 words/row: 16B-aligned b128 LDS targets
#define TILE_WORDS (TL * PADW)

typedef __attribute__((ext_vector_type(2))) float v2f;
typedef __attribute__((ext_vector_type(8))) float v8f;

// Vector-of-4-int type matching the async builtin's parameter type.
typedef int b128_t __attribute__((vector_size(16)));
typedef __attribute__((address_space(1))) b128_t* gptr128_t;
typedef __attribute__((address_space(3))) b128_t* lptr128_t;

// ---- CDNA5 async copy (global -> LDS), ASYNCcnt-tracked ---------------------
#if defined(__has_builtin) && __has_builtin(__builtin_amdgcn_global_load_async_to_lds_b128)
static __device__ __forceinline__ void async_cp16(const float* g, float* l) {
    __builtin_amdgcn_global_load_async_to_lds_b128(
        (gptr128_t)(g), (lptr128_t)(l), 0, 0);
}
#else
static __device__ __forceinline__ void async_cp16(const float* g, float* l) {
    unsigned lofs = (unsigned)(size_t)(__attribute__((address_space(3))) char*)l;
    asm volatile("global_load_async_to_lds_b128 %0, %1, off"
                 :: "v"(lofs), "v"(g) : "memory");
}
#endif

#if defined(__has_builtin) && __has_builtin(__builtin_amdgcn_s_wait_asynccnt)
#define WAIT_ASYNC(n) __builtin_amdgcn_s_wait_asynccnt(n)
#else
#define WAIT_ASYNC(n) asm volatile("s_wait_asynccnt %0" :: "i"(n) : "memory")
#endif

// ---------------------------------------------------------------------------
// Kernel A: q = cross_input @ W   ([N,128] x [128,128] row-major W[e][d])
// One wave per 16x16 tile, V_WMMA_F32_16X16X4_F32, K-loop of 32 steps.
// ---------------------------------------------------------------------------
__global__ __launch_bounds__(32) void qgemm_wmma(
    const float* __restrict__ cross,   // [N, DIM]
    const float* __restrict__ W,       // [DIM, DIM]  W[e][d]
    float* __restrict__ q)             // [N, DIM]
{
    const int nBase   = blockIdx.x * 16;
    const int dBase   = blockIdx.y * 16;
    const int lane    = threadIdx.x;
    const int halfsel = lane >> 4;
    const int lid     = lane & 15;

    v8f c = {};
    for (int k0 = 0; k0 < DIM; k0 += 4) {
        const float* ap = &cross[(nBase + lid) * DIM + k0 + 2 * halfsel];
        v2f a; a.x = ap[0]; a.y = ap[1];
        const float* bp = &W[(k0 + halfsel) * DIM + dBase + lid];
        v2f b; b.x = bp[0]; b.y = bp[2 * DIM];
        c = __builtin_amdgcn_wmma_f32_16x16x4_f32(
                false, a, false, b, (short)0, c, false, false);
    }
#pragma unroll
    for (int j = 0; j < 8; ++j)
        q[(nBase + j + 8 * halfsel) * DIM + dBase + lid] = c[j];
}

// ---------------------------------------------------------------------------
// Kernel B: fused scores + online softmax + ctx + (ctx.v)/s, one block per n.
// Double-buffered GLOBAL_LOAD_ASYNC_TO_LDS_B128 pipeline: next tile's HBM
// fetch overlaps current tile's compute. Single pass over input_seq (1.07 GB).
// ---------------------------------------------------------------------------
__global__ __launch_bounds__(256) void attn_fused(
    const float* __restrict__ x,      // [N, L, DIM]
    const float* __restrict__ qg,     // [N, DIM]
    const float* __restrict__ v,      // [DIM]
    float* __restrict__ out,          // [N]
    int L)
{
    const int n   = blockIdx.x;
    const int tid = threadIdx.x;

    __shared__ float sxd[2 * TILE_WORDS];   // double-buffered tiles (padded rows)
    __shared__ float ssc[TL];
    __shared__ float sq[DIM];
    __shared__ float spart[4][TL];
    __shared__ float sacc[DIM];
    __shared__ float sfin[DIM];
    __shared__ float sredmax;
    __shared__ float sredsum;

    if (tid < DIM) sq[tid] = qg[n * DIM + tid];

    const int d       = tid & (DIM - 1);
    const int half    = tid >> 7;
    const int row     = tid & (TL - 1);
    const int quarter = tid >> 6;

    float m = -INFINITY, s = 0.f, acc = 0.f;

    const float* xb = x + (size_t)n * L * DIM;
    const int ntiles = L / TL;

    // prologue: async-fetch tile 0 into buffer 0 (8 b128 per thread)
    {
        const float* src = xb;
#pragma unroll
        for (int i = 0; i < 8; ++i) {
            int vi = i * 256 + tid;             // 0..2047 float4s
            int r  = vi >> 5;                   // 32 float4 per row
            int c  = (vi & 31) << 2;
            async_cp16(src + r * DIM + c, &sxd[r * PADW + c]);
        }
    }

    for (int t = 0; t < ntiles; ++t) {
        const float* cur = &sxd[(t & 1) * TILE_WORDS];

        if (t + 1 < ntiles) {                   // issue next tile, then wait for current
            const float* src = xb + (size_t)(t + 1) * TL * DIM;
            float* nbuf = &sxd[((t + 1) & 1) * TILE_WORDS];
#pragma unroll
            for (int i = 0; i < 8; ++i) {
                int vi = i * 256 + tid;
                int r  = vi >> 5;
                int c  = (vi & 31) << 2;
                async_cp16(src + r * DIM + c, &nbuf[r * PADW + c]);
            }
            WAIT_ASYNC(8);                      // current tile's 8 done (in-order); next 8 in flight
        } else {
            WAIT_ASYNC(0);
        }
        __syncthreads();                        // whole tile visible to all waves

        // partial dots: 4 threads per row, each covers 32 d's
        float part = 0.f;
        {
            const float* xr = &cur[row * PADW + quarter * 32];
            const float* qr = &sq[quarter * 32];
#pragma unroll
            for (int j = 0; j < 32; ++j) part = fmaf(xr[j], qr[j], part);
        }
        spart[quarter][row] = part;
        __syncthreads();
        if (tid < TL)
            ssc[tid] = spart[0][tid] + spart[1][tid] + spart[2][tid] + spart[3][tid];
        __syncthreads();

        // tile max (wave 0)
        if (tid < 32) {
            float mx = fmaxf(ssc[tid], ssc[tid + 32]);
            for (int off = 16; off > 0; off >>= 1)
                mx = fmaxf(mx, __shfl_xor(mx, off, 32));
            if (tid == 0) sredmax = mx;
        }
        __syncthreads();

        float new_m  = fmaxf(m, sredmax);
        float factor = __expf(m - new_m);       // 0 on first tile (m = -inf)
        m = new_m;
        acc *= factor;
        if (tid < TL) ssc[tid] = __expf(ssc[tid] - new_m);
        __syncthreads();

        // tile sum (wave 0)
        if (tid < 32) {
            float sm = ssc[tid] + ssc[tid + 32];
            for (int off = 16; off > 0; off >>= 1)
                sm += __shfl_xor(sm, off, 32);
            if (tid == 0) sredsum = sm;
        }
        __syncthreads();
        s = s * factor + sredsum;

        // ctx accumulation: thread owns column d, half of the rows
        {
            const int lbase = half * 32;
#pragma unroll 4
            for (int i = 0; i < 32; ++i) {
                int l = lbase + i;
                acc = fmaf(ssc[l], cur[l * PADW + d], acc);
            }
        }
        __syncthreads();                        // done reading cur before t+2 overwrites it
    }

    if (half == 1) sacc[d] = acc;               // merge row-halves
    __syncthreads();
    if (half == 0) sfin[d] = (acc + sacc[d]) * v[d];
    __syncthreads();
    if (tid < 32) {
        float sm = sfin[tid] + sfin[tid + 32] + sfin[tid + 64] + sfin[tid + 96];
        for (int off = 16; off > 0; off >>= 1)
            sm += __shfl_xor(sm, off, 32);
        if (tid == 0) out[n] = sm / s;          // normalize by softmax sum
    }
}

// ---------------------------------------------------------------------------
extern "C" void kernel_launch(void* const* d_in, const int* in_sizes, int n_in,
                              void* d_out, int out_size, void* d_ws, size_t ws_size,
                              hipStream_t stream)
{
    (void)n_in; (void)out_size; (void)ws_size;
    const float* x     = (const float*)d_in[0];   // input_seq [N,L,D]
    const float* cross = (const float*)d_in[1];   // cross_input [N,D]
    const float* W     = (const float*)d_in[2];   // W [D,D]
    const float* v     = (const float*)d_in[3];   // v [D]
    float* out = (float*)d_out;                   // [N,1] -> N floats
    float* q   = (float*)d_ws;                    // [N,D] scratch: 512 KB

    const int N = in_sizes[1] / DIM;
    const int L = in_sizes[0] / (N * DIM);

    qgemm_wmma<<<dim3(N / 16, DIM / 16), 32, 0, stream>>>(cross, W, q);
    attn_fused<<<N, 256, 0, stream>>>(x, q, v, out, L);
}